// BaseLUT_2886218023576
// MI455X (gfx1250) — compile-verified
//
#include <hip/hip_runtime.h>
#include <stdint.h>

// -------- problem constants --------
#define IMG_H 128
#define IMG_W 128
#define NIMG  12          // 4 * 3 (n*c planes)
#define TS    16          // tile of base pixels per block
#define TW    (TS + 4)    // 20x20 stencil tile (halo 2 on each side, window-shifted)
#define L17   17

typedef __attribute__((ext_vector_type(4))) unsigned int u32x4;
typedef __attribute__((ext_vector_type(4))) int          i32x4;
typedef __attribute__((ext_vector_type(8))) int          i32x8;

#if defined(__has_builtin)
#if __has_builtin(__builtin_amdgcn_tensor_load_to_lds) && __has_builtin(__builtin_amdgcn_s_wait_tensorcnt)
#define USE_TDM 1
#endif
#endif
#ifndef USE_TDM
#define USE_TDM 0
#endif

// ---------------------------------------------------------------------------
// CDNA5 Tensor Data Mover: async DMA of a 2D f32 tile (tile_w x tile_h) from a
// row-major tensor (row stride = stride elems) into LDS at byte offset lds_off.
// Descriptor fields per CDNA5 ISA ch.8 (D# group0 128b, group1 256b).
// ---------------------------------------------------------------------------
#if USE_TDM
__device__ __forceinline__ void tdm_load_tile_f32(const float* gsrc, uint32_t lds_off,
                                                  int rem_w, int rem_h,
                                                  int tile_w, int tile_h, int stride) {
  uint64_t ga = (uint64_t)(uintptr_t)gsrc;
  u32x4 g0;
  g0.x = 1u;                                              // count=1, user mode
  g0.y = lds_off;                                         // lds_addr [63:32]
  g0.z = (uint32_t)ga;                                    // global_addr [95:64]
  g0.w = (uint32_t)((ga >> 32) & 0x01FFFFFFull) | (2u << 30);  // addr hi | type=2
  i32x8 g1;
  g1.s0 = 0x00020000;                                     // data_size=2 (4 bytes)
  g1.s1 = (rem_w  & 0xFFFF) << 16;                        // tensor_dim0 lo16
  g1.s2 = (rem_h  & 0xFFFF) << 16;                        // tensor_dim1 lo16
  g1.s3 = (tile_w & 0xFFFF) << 16;                        // tile_dim0
  g1.s4 = (tile_h & 0xFFFF);                              // tile_dim1 (tile_dim2=0)
  g1.s5 = stride;                                         // tensor_dim0_stride lo32
  g1.s6 = 0;
  g1.s7 = 0;
  i32x4 z4 = (i32x4)0;
#if __clang_major__ >= 23
  i32x8 z8 = (i32x8)0;
  __builtin_amdgcn_tensor_load_to_lds(g0, g1, z4, z4, z8, 0);
#else
  __builtin_amdgcn_tensor_load_to_lds(g0, g1, z4, z4, 0);
#endif
  __builtin_amdgcn_s_wait_tensorcnt((short)0);            // s_wait_tensorcnt 0
}
#endif

// -------- LUT quantization: clip(rint(w*127), -127, 127), RNE like jnp.round
__device__ __forceinline__ float lutT(float w) {
  return fminf(fmaxf(rintf(w * 127.0f), -127.0f), 127.0f);
}

// p_{b3 b2 b1 b0}: slot0=Ma(+b3), slot1 = b2 ? (Ma+1) : Mb  (faithful: Na, not Nb),
// slot2=Mc(+b1), slot3=Md(+b0)
__device__ __forceinline__ int lut_idx(int Ma, int Mb, int Mc, int Md, int code) {
  int a = Ma + ((code >> 3) & 1);
  int b = (code & 4) ? (Ma + 1) : Mb;
  int c = Mc + ((code >> 1) & 1);
  int d = Md + (code & 1);
  return ((a * L17 + b) * L17 + c) * L17 + d;
}

struct Simplex { float w0, w1, w2, w3, w4; int i0, i1, i2, i3, i4; };

// 24-case 4D simplex selection, exact cascade of the reference (incl. dead c10).
__device__ __forceinline__ Simplex simplex_setup(float va, float vb, float vc, float vd) {
  int   Ma = (int)floorf(va * 0.0625f); float La = va - 16.0f * (float)Ma;
  int   Mb = (int)floorf(vb * 0.0625f); float Lb = vb - 16.0f * (float)Mb;
  int   Mc = (int)floorf(vc * 0.0625f); float Lc = vc - 16.0f * (float)Mc;
  int   Md = (int)floorf(vd * 0.0625f); float Ld = vd - 16.0f * (float)Md;
  bool Lab = La > Lb, Lac = La > Lc, Lad = La > Ld;
  bool Lbc = Lb > Lc, Lbd = Lb > Ld, Lcd = Lc > Ld;
  float l1, l2, l3, l4; int c2, c3, c4;
  if (Lab) {
    if (Lbc) {
      if      (Lcd) { l1=La; l2=Lb; l3=Lc; l4=Ld; c2=0x8; c3=0xC; c4=0xE; }  // c1
      else if (Lbd) { l1=La; l2=Lb; l3=Ld; l4=Lc; c2=0x8; c3=0xC; c4=0xD; }  // c2
      else if (Lad) { l1=La; l2=Ld; l3=Lb; l4=Lc; c2=0x8; c3=0x9; c4=0xD; }  // c3
      else          { l1=Ld; l2=La; l3=Lb; l4=Lc; c2=0x1; c3=0x9; c4=0xD; }  // c4
    } else if (Lac) {
      if      (Lbd) { l1=La; l2=Lc; l3=Lb; l4=Ld; c2=0x8; c3=0xA; c4=0xE; }  // c5
      else if (Lcd) { l1=La; l2=Lc; l3=Ld; l4=Lb; c2=0x8; c3=0xA; c4=0xB; }  // c6
      else if (Lad) { l1=La; l2=Ld; l3=Lc; l4=Lb; c2=0x8; c3=0x9; c4=0xB; }  // c7
      else          { l1=Ld; l2=La; l3=Lc; l4=Lb; c2=0x1; c3=0x9; c4=0xB; }  // c8
    } else {
      if      (Lad) { l1=Lc; l2=La; l3=Lb; l4=Ld; c2=0x2; c3=0xA; c4=0xE; }  // c9 (c10 empty)
      else if (Lcd) { l1=Lc; l2=Ld; l3=La; l4=Lb; c2=0x2; c3=0x3; c4=0xB; }  // c11
      else          { l1=Ld; l2=Lc; l3=La; l4=Lb; c2=0x1; c3=0x3; c4=0xB; }  // c12
    }
  } else {
    if (Lac) {
      if      (Lcd) { l1=Lb; l2=La; l3=Lc; l4=Ld; c2=0x4; c3=0xC; c4=0xE; }  // c13
      else if (Lad) { l1=Lb; l2=La; l3=Ld; l4=Lc; c2=0x4; c3=0xC; c4=0xD; }  // c14
      else if (Lbd) { l1=Lb; l2=Ld; l3=La; l4=Lc; c2=0x4; c3=0x5; c4=0xD; }  // c15
      else          { l1=Ld; l2=Lb; l3=La; l4=Lc; c2=0x1; c3=0x5; c4=0xD; }  // c16
    } else if (Lbc) {
      if      (Lad) { l1=Lb; l2=Lc; l3=La; l4=Ld; c2=0x4; c3=0x6; c4=0xE; }  // c17
      else if (Lcd) { l1=Lb; l2=Lc; l3=Ld; l4=La; c2=0x4; c3=0x6; c4=0x7; }  // c18
      else if (Lbd) { l1=Lb; l2=Ld; l3=Lc; l4=La; c2=0x4; c3=0x5; c4=0x7; }  // c19
      else          { l1=Ld; l2=Lb; l3=Lc; l4=La; c2=0x1; c3=0x5; c4=0x7; }  // c20
    } else {
      if      (Lad) { l1=Lc; l2=Lb; l3=La; l4=Ld; c2=0x2; c3=0x6; c4=0xE; }  // c21
      else if (Lbd) { l1=Lc; l2=Lb; l3=Ld; l4=La; c2=0x2; c3=0x6; c4=0x7; }  // c22
      else if (Lcd) { l1=Lc; l2=Ld; l3=Lb; l4=La; c2=0x2; c3=0x3; c4=0x7; }  // c23
      else          { l1=Ld; l2=Lc; l3=Lb; l4=La; c2=0x1; c3=0x3; c4=0x7; }  // c24
    }
  }
  Simplex s;
  s.w0 = 16.0f - l1; s.w2 = l1 - l2; s.w3 = l2 - l3; s.w4 = l3 - l4; s.w1 = l4;
  s.i0 = lut_idx(Ma, Mb, Mc, Md, 0x0);
  s.i1 = lut_idx(Ma, Mb, Mc, Md, 0xF);
  s.i2 = lut_idx(Ma, Mb, Mc, Md, c2);
  s.i3 = lut_idx(Ma, Mb, Mc, Md, c3);
  s.i4 = lut_idx(Ma, Mb, Mc, Md, c4);
  return s;
}

// mode tap offsets in the rotated frame: s, d, y
constexpr int OFFS[3][4][2] = {
  {{0,0},{0,1},{1,0},{1,1}},
  {{0,0},{0,2},{2,0},{2,2}},
  {{0,0},{1,1},{1,2},{2,1}}};

// rotate offsets into original frame; r in [0,4)
__device__ __forceinline__ void rot_off(int r, int dy, int dx, int& oy, int& ox) {
  if      (r == 0) { oy =  dy; ox =  dx; }
  else if (r == 1) { oy =  dx; ox = -dy; }
  else if (r == 2) { oy = -dy; ox = -dx; }
  else             { oy = -dx; ox =  dy; }
}

// block-cooperative tile stage (TDM on gfx1250, fallback plain loads)
__device__ __forceinline__ void stage_tile(const float* src, float* tile,
                                           int row0, int col0, int tid) {
#if USE_TDM
  if (tid < 32) {  // wave 0 issues one 2D DMA for the whole block
    tdm_load_tile_f32(src + row0 * IMG_W + col0,
                      (uint32_t)(uintptr_t)(void*)tile,
                      IMG_W - col0, IMG_H - row0, TW, TW, IMG_W);
  }
#else
  for (int t = tid; t < TW * TW; t += 256)
    tile[t] = src[(row0 + t / TW) * IMG_W + (col0 + t % TW)];
#endif
  __syncthreads();
}

// ------------------------- stage 1: count=1, upscale=1 ----------------------
__global__ __launch_bounds__(256) void lut_stage1(const float* __restrict__ x,
                                                  const float* __restrict__ lutS,
                                                  const float* __restrict__ lutD,
                                                  const float* __restrict__ lutY,
                                                  float* __restrict__ img1) {
  __shared__ float tile[TW * TW];
  const int img = blockIdx.z;
  const int i0 = blockIdx.y * TS, j0 = blockIdx.x * TS;
  const int tx = threadIdx.x, ty = threadIdx.y, tid = ty * TS + tx;
  const int row0 = min(max(i0 - 2, 0), IMG_H - TW);
  const int col0 = min(max(j0 - 2, 0), IMG_W - TW);
  const float* src = x + (size_t)img * (IMG_H * IMG_W);

  stage_tile(src, tile, row0, col0, tid);
  // img = clip(x,0,1)*255, done once in LDS
  for (int t = tid; t < TW * TW; t += 256)
    tile[t] = fminf(fmaxf(tile[t], 0.0f), 1.0f) * 255.0f;
  __syncthreads();

  const int i = i0 + ty, j = j0 + tx;
  const float* luts[3] = {lutS, lutD, lutY};
  float pred = 0.0f;
  for (int m = 0; m < 3; m++) {
    const float* lut = luts[m];
#pragma unroll
    for (int r = 0; r < 4; r++) {
      float v[4];
#pragma unroll
      for (int t = 0; t < 4; t++) {
        int oy, ox; rot_off(r, OFFS[m][t][0], OFFS[m][t][1], oy, ox);
        int ti = min(max(i + oy, 0), IMG_H - 1) - row0;
        int tj = min(max(j + ox, 0), IMG_W - 1) - col0;
        v[t] = tile[ti * TW + tj];
      }
      Simplex s = simplex_setup(v[0], v[1], v[2], v[3]);
      pred += s.w0 * lutT(lut[s.i0]) + s.w2 * lutT(lut[s.i2]) + s.w3 * lutT(lut[s.i3])
            + s.w4 * lutT(lut[s.i4]) + s.w1 * lutT(lut[s.i1]);
    }
  }
  img1[(size_t)img * (IMG_H * IMG_W) + i * IMG_W + j] =
      rintf(fminf(fmaxf(pred / 12.0f + 127.0f, 0.0f), 255.0f));
}

// ------------------------- stage 2: count=16, upscale=4 ---------------------
__global__ __launch_bounds__(256) void lut_stage2(const float* __restrict__ img1,
                                                  const float* __restrict__ lutS,
                                                  const float* __restrict__ lutD,
                                                  const float* __restrict__ lutY,
                                                  float* __restrict__ out) {
  __shared__ float tile[TW * TW];
  const int img = blockIdx.z;
  const int i0 = blockIdx.y * TS, j0 = blockIdx.x * TS;
  const int tx = threadIdx.x, ty = threadIdx.y, tid = ty * TS + tx;
  const int row0 = min(max(i0 - 2, 0), IMG_H - TW);
  const int col0 = min(max(j0 - 2, 0), IMG_W - TW);
  const float* src = img1 + (size_t)img * (IMG_H * IMG_W);

  stage_tile(src, tile, row0, col0, tid);

  const int i = i0 + ty, j = j0 + tx;
  const float* luts[3] = {lutS, lutD, lutY};
  float acc[16];
#pragma unroll
  for (int k = 0; k < 16; k++) acc[k] = 0.0f;

  for (int m = 0; m < 3; m++) {
    const float* lut = luts[m];
#pragma unroll
    for (int r = 0; r < 4; r++) {
      float v[4];
#pragma unroll
      for (int t = 0; t < 4; t++) {
        int oy, ox; rot_off(r, OFFS[m][t][0], OFFS[m][t][1], oy, ox);
        int ti = min(max(i + oy, 0), IMG_H - 1) - row0;
        int tj = min(max(j + ox, 0), IMG_W - 1) - col0;
        v[t] = tile[ti * TW + tj];
      }
      Simplex s = simplex_setup(v[0], v[1], v[2], v[3]);
      const float* r0 = lut + (size_t)s.i0 * 16;
      const float* r2 = lut + (size_t)s.i2 * 16;
      const float* r3 = lut + (size_t)s.i3 * 16;
      const float* r4 = lut + (size_t)s.i4 * 16;
      const float* r1 = lut + (size_t)s.i1 * 16;
#pragma unroll
      for (int k = 0; k < 16; k++) {
        float val = s.w0 * lutT(r0[k]) + s.w2 * lutT(r2[k]) + s.w3 * lutT(r3[k])
                  + s.w4 * lutT(r4[k]) + s.w1 * lutT(r1[k]);
        // rotate sub-pixel index back to original orientation
        int kr = k >> 2, kc = k & 3, u, v2;
        if      (r == 0) { u = kr;     v2 = kc;     }
        else if (r == 1) { u = kc;     v2 = 3 - kr; }
        else if (r == 2) { u = 3 - kr; v2 = 3 - kc; }
        else             { u = 3 - kc; v2 = kr;     }
        acc[u * 4 + v2] += val;
      }
    }
  }

  float* op = out + (size_t)img * (IMG_H * 4) * (IMG_W * 4);
#pragma unroll
  for (int u = 0; u < 4; u++)
#pragma unroll
    for (int v2 = 0; v2 < 4; v2++)
      op[(size_t)(i * 4 + u) * (IMG_W * 4) + (j * 4 + v2)] =
          rintf(fminf(fmaxf(acc[u * 4 + v2] / 3.0f, 0.0f), 255.0f)) / 255.0f;
}

// ---------------------------------------------------------------------------
extern "C" void kernel_launch(void* const* d_in, const int* in_sizes, int n_in,
                              void* d_out, int out_size, void* d_ws, size_t ws_size,
                              hipStream_t stream) {
  (void)in_sizes; (void)n_in; (void)out_size; (void)ws_size;
  const float* x    = (const float*)d_in[0];
  const float* w1s  = (const float*)d_in[1];
  const float* w1d  = (const float*)d_in[2];
  const float* w1y  = (const float*)d_in[3];
  const float* w2s  = (const float*)d_in[4];
  const float* w2d  = (const float*)d_in[5];
  const float* w2y  = (const float*)d_in[6];
  float* img1 = (float*)d_ws;                 // 12*128*128 f32 = 768 KB scratch
  float* out  = (float*)d_out;                // 4*3*512*512 f32

  dim3 blk(TS, TS, 1);
  dim3 grd(IMG_W / TS, IMG_H / TS, NIMG);
  lut_stage1<<<grd, blk, 0, stream>>>(x, w1s, w1d, w1y, img1);
  lut_stage2<<<grd, blk, 0, stream>>>(img1, w2s, w2d, w2y, out);
}